// projection_Measurement_85813446574645
// MI455X (gfx1250) — compile-verified
//
#include <hip/hip_runtime.h>

#define D 4096
#define B 8192
#define KSPLIT 4
#define KCHUNK (D / KSPLIT)   // 1024

typedef float v2f __attribute__((ext_vector_type(2)));
typedef float v8f __attribute__((ext_vector_type(8)));

// ---------------- Phase 1: normalize the two projector columns ----------------
__global__ void __launch_bounds__(256)
k_normalize(const float* __restrict__ proj,
            float* __restrict__ prn, float* __restrict__ pin) {
    __shared__ float red0[8], red1[8];
    const int t = threadIdx.x;
    float s0 = 0.f, s1 = 0.f;
    for (int i = t; i < D; i += 256) {
        const float a = proj[i];
        const float b = proj[D + i];
        s0 += a * a;
        s1 += b * b;
    }
    for (int off = 16; off > 0; off >>= 1) {
        s0 += __shfl_down(s0, off, 32);
        s1 += __shfl_down(s1, off, 32);
    }
    const int wave = t >> 5, lane = t & 31;
    if (lane == 0) { red0[wave] = s0; red1[wave] = s1; }
    __syncthreads();
    if (t == 0) {
        float a = 0.f, b = 0.f;
        for (int w = 0; w < 8; ++w) { a += red0[w]; b += red1[w]; }
        red0[0] = 1.0f / sqrtf(a);
        red1[0] = 1.0f / sqrtf(b);
    }
    __syncthreads();
    const float ir = red0[0], ii = red1[0];
    for (int i = t; i < D; i += 256) {
        prn[i] = proj[i] * ir;
        pin[i] = proj[D + i] * ii;
    }
}

// ---------------- Phase 2: 4 GEMV dot products via V_WMMA_F32_16X16X4_F32 ----
// Block = 64 threads (2 waves). wave0: v_real rows, wave1: v_imag rows.
// blockIdx.x -> 16 batch rows, blockIdx.y -> K-chunk of size KCHUNK.
// B-operand columns: n==0 carries p_r_hat, n==1 carries p_i_hat, rest zero
// (selection done branch-free with float masks so EXEC stays all-1s and the
//  hot loop has no control flow).
// part layout: part[(chunk*4 + q)*B + row], q: 0=vr.pr(a) 1=vr.pi(d) 2=vi.pr(e) 3=vi.pi(c)
__global__ void __launch_bounds__(64)
k_dots_wmma(const float* __restrict__ vr, const float* __restrict__ vi,
            const float* __restrict__ prn, const float* __restrict__ pin,
            float* __restrict__ part) {
    __shared__ float s_pr[KCHUNK];
    __shared__ float s_pi[KCHUNK];
    __shared__ float s_tile[2][16 * 68];   // 68-float row stride: bank pad + 16B align

    const int wave  = threadIdx.x >> 5;
    const int lane  = threadIdx.x & 31;
    const int half  = lane >> 4;           // selects K pair base (+0 / +2)
    const int n     = lane & 15;           // N column (B/C) == M row (A fragment)
    const int row0  = blockIdx.x * 16;
    const int chunk = blockIdx.y;
    const int kg0   = chunk * KCHUNK;

    // branch-free B-column masks (computed once)
    const float m_r = (n == 0) ? 1.0f : 0.0f;
    const float m_i = (n == 1) ? 1.0f : 0.0f;

    // stage this K-chunk of the normalized projector in LDS (broadcast reads later)
    for (int i = threadIdx.x; i < KCHUNK; i += 64) {
        s_pr[i] = prn[kg0 + i];
        s_pi[i] = pin[kg0 + i];
    }
    __syncthreads();

    const float* __restrict__ V = wave ? vi : vr;
    float* tile = &s_tile[wave][0];

    v8f acc = {0.f, 0.f, 0.f, 0.f, 0.f, 0.f, 0.f, 0.f};

    for (int k0 = 0; k0 < KCHUNK; k0 += 64) {
        // cooperative, fully-coalesced 16x64 f32 tile load (8 x b128 per lane)
        #pragma unroll
        for (int i = 0; i < 8; ++i) {
            const int id = lane + i * 32;
            const int r  = id >> 4;     // 0..15 (row)
            const int c4 = id & 15;     // 0..15 (float4 column)
            const float4 g = *reinterpret_cast<const float4*>(
                V + (size_t)(row0 + r) * D + kg0 + k0 + c4 * 4);
            *reinterpret_cast<float4*>(tile + r * 68 + c4 * 4) = g;
        }
        // 16 WMMA steps cover K=64 of this tile; no branches in this loop.
        #pragma unroll
        for (int kk = 0; kk < 16; ++kk) {
            const int kb = kk * 4 + 2 * half;     // K offset within tile (even -> 8B aligned)
            // A fragment (16x4 f32): lane m=lane&15 holds A[m][kb], A[m][kb+1]
            const float2 av = *reinterpret_cast<const float2*>(tile + n * 68 + kb);
            // broadcast projector pairs (same address across each half-wave)
            const int kl = k0 + kb;
            const float2 prv = *reinterpret_cast<const float2*>(s_pr + kl);
            const float2 piv = *reinterpret_cast<const float2*>(s_pi + kl);
            v2f a; a.x = av.x; a.y = av.y;
            v2f b;
            b.x = m_r * prv.x + m_i * piv.x;
            b.y = m_r * prv.y + m_i * piv.y;
            acc = __builtin_amdgcn_wmma_f32_16x16x4_f32(
                false, a, false, b, (short)0, acc, false, false);
        }
    }

    // C layout: VGPR j, lanes 0-15 -> M=j, lanes 16-31 -> M=j+8, N=lane&15.
    // Column 0 = dot with p_r_hat, column 1 = dot with p_i_hat.
    if (n < 2) {
        const int q = wave * 2 + n;
        float* o = part + (size_t)(chunk * 4 + q) * B + row0 + 8 * half;
        #pragma unroll
        for (int j = 0; j < 8; ++j) o[j] = acc[j];
    }
}

// ---------------- Phase 2b: deterministic K-split reduction ------------------
__global__ void __launch_bounds__(256)
k_reduce_parts(const float* __restrict__ part, float* __restrict__ coef) {
    const int i = blockIdx.x * 256 + threadIdx.x;   // over 4*B coefficients
    if (i < 4 * B) {
        float s = 0.f;
        #pragma unroll
        for (int c = 0; c < KSPLIT; ++c) s += part[(size_t)c * 4 * B + i];
        coef[i] = s;
    }
}

// ---------------- Phase 3: rank-2 outer-product writeback --------------------
__global__ void __launch_bounds__(256)
k_combine(const float* __restrict__ prn, const float* __restrict__ pin,
          const float* __restrict__ coef,
          float* __restrict__ outR, float* __restrict__ outI) {
    const int b  = blockIdx.x;
    const float ca = coef[0 * B + b];   // vr . p_r
    const float cd = coef[1 * B + b];   // vr . p_i
    const float ce = coef[2 * B + b];   // vi . p_r
    const float cc = coef[3 * B + b];   // vi . p_i
    float* oR = outR + (size_t)b * D;
    float* oI = outI + (size_t)b * D;
    for (int j = threadIdx.x * 4; j < D; j += 256 * 4) {
        const float4 pr = *reinterpret_cast<const float4*>(prn + j);
        const float4 pi = *reinterpret_cast<const float4*>(pin + j);
        float4 r, im;
        r.x  = ca * pr.x - cc * pi.x;
        r.y  = ca * pr.y - cc * pi.y;
        r.z  = ca * pr.z - cc * pi.z;
        r.w  = ca * pr.w - cc * pi.w;
        im.x = cd * pi.x + ce * pr.x;
        im.y = cd * pi.y + ce * pr.y;
        im.z = cd * pi.z + ce * pr.z;
        im.w = cd * pi.w + ce * pr.w;
        *reinterpret_cast<float4*>(oR + j) = r;
        *reinterpret_cast<float4*>(oI + j) = im;
    }
}

extern "C" void kernel_launch(void* const* d_in, const int* in_sizes, int n_in,
                              void* d_out, int out_size, void* d_ws, size_t ws_size,
                              hipStream_t stream) {
    const float* vr   = (const float*)d_in[0];   // v_real_avg [B, D]
    const float* vi   = (const float*)d_in[1];   // v_imag_avg [B, D]
    const float* proj = (const float*)d_in[2];   // projector  [2, D, 1]

    float* w    = (float*)d_ws;
    float* prn  = w;                         // [D]
    float* pin  = w + D;                     // [D]
    float* coef = w + 2 * D;                 // [4*B]
    float* part = w + 2 * D + 4 * B;         // [KSPLIT*4*B]

    float* outR = (float*)d_out;             // Pv_real [B*D]
    float* outI = outR + (size_t)B * D;      // Pv_imag [B*D]

    k_normalize<<<1, 256, 0, stream>>>(proj, prn, pin);
    k_dots_wmma<<<dim3(B / 16, KSPLIT), 64, 0, stream>>>(vr, vi, prn, pin, part);
    k_reduce_parts<<<(4 * B + 255) / 256, 256, 0, stream>>>(part, coef);
    k_combine<<<B, 256, 0, stream>>>(prn, pin, coef, outR, outI);
}